// Network_28295244546215
// MI455X (gfx1250) — compile-verified
//
#include <hip/hip_runtime.h>
#include <math.h>

// CDNA5 / gfx1250, wave32. Streaming fp32 matvec through v_wmma_f32_16x16x4_f32.

typedef __attribute__((ext_vector_type(2))) float v2f;
typedef __attribute__((ext_vector_type(8))) float v8f;

#define NN 16384
#define THRESH 15.0f
#define INV_SQRT2 0.70710678118654752f

#define BLOCK 256            // 8 wave32
#define WAVES 8
#define ROWS 16              // rows per block (one WMMA M-tile)
#define KCHUNK (NN / WAVES)  // 2048 K per wave

__global__ __launch_bounds__(BLOCK) void rnn_step_wmma(
    const float* __restrict__ rates,
    const float* __restrict__ noise,
    const float* __restrict__ W,
    const float* __restrict__ bias,
    const float* __restrict__ exp_dt_tau,
    const float* __restrict__ dt_tau,
    float* __restrict__ out)
{
    // Exactly 64 KB of LDS. The first 128 floats are reused as the
    // per-wave partial-sum buffer after the K loop (guarded by barriers).
    __shared__ float sh_rates[NN];

    const int tid  = threadIdx.x;
    const int wave = tid >> 5;
    const int lane = tid & 31;
    const int row0 = blockIdx.x * ROWS;

    // ---- cooperative stage of the rates vector into LDS (float4) ----
    {
        const float4* __restrict__ r4 = (const float4*)rates;
        float4* s4 = (float4*)sh_rates;
        #pragma unroll
        for (int i = 0; i < (NN / 4) / BLOCK; ++i)
            s4[tid + i * BLOCK] = r4[tid + i * BLOCK];
    }
    __syncthreads();

    // ---- WMMA streaming matvec ----
    // A (16x4 f32): lane&15 = M (row), lane-half selects K offset {0,2},
    //               two VGPRs hold K and K+1 -> one contiguous float2 per lane.
    // B (4x16 f32): same slot->K mapping filled with rates, identical across
    //               all N columns -> every lane of D holds the row partial.
    const int m      = lane & 15;
    const int coloff = (lane >> 4) << 1;      // 0 or 2
    const int kbase  = wave * KCHUNK;

    const float* __restrict__ wptr = W + (size_t)(row0 + m) * NN + kbase + coloff;
    const float* sptr = sh_rates + kbase + coloff;

    v8f acc = {};
    #pragma unroll 8
    for (int k = 0; k < KCHUNK; k += 4) {
        v2f a = *(const v2f*)(wptr + k);      // global_load_b64 of W
        v2f b = *(const v2f*)(sptr + k);      // ds_load_b64 broadcast of rates
        acc = __builtin_amdgcn_wmma_f32_16x16x4_f32(
                  false, a, false, b, (short)0, acc, false, false);
    }

    // ---- reduce wave partials across the 8 K-slices ----
    __syncthreads();                          // all waves done reading sh_rates
    float* sh_part = sh_rates;                // reuse: WAVES*ROWS = 128 floats

    // All D columns are equal; lanes 0 and 16 carry M=0..7 / M=8..15 (VGPR v).
    if ((lane & 15) == 0) {
        const int mbase = (lane >> 4) * 8;
        #pragma unroll
        for (int v = 0; v < 8; ++v)
            sh_part[wave * ROWS + mbase + v] = acc[v];
    }
    __syncthreads();

    // ---- epilogue: bias + noise, erf activation, decay update ----
    if (tid < ROWS) {
        float y = 0.0f;
        #pragma unroll
        for (int w = 0; w < WAVES; ++w)
            y += sh_part[w * ROWS + tid];

        const int row = row0 + tid;
        const float x   = y + bias[row] + noise[row];
        const float act = THRESH * 0.5f * (1.0f + erff(x * INV_SQRT2));
        out[row] = rates[row] * exp_dt_tau[row] + dt_tau[row] * act;
    }
}

extern "C" void kernel_launch(void* const* d_in, const int* in_sizes, int n_in,
                              void* d_out, int out_size, void* d_ws, size_t ws_size,
                              hipStream_t stream) {
    (void)in_sizes; (void)n_in; (void)d_ws; (void)ws_size; (void)out_size;

    const float* rates      = (const float*)d_in[0];
    const float* noise      = (const float*)d_in[1];
    const float* W          = (const float*)d_in[2];
    const float* bias       = (const float*)d_in[3];
    const float* exp_dt_tau = (const float*)d_in[4];
    const float* dt_tau     = (const float*)d_in[5];
    float* out              = (float*)d_out;

    dim3 grid(NN / ROWS);   // 1024 blocks
    dim3 block(BLOCK);      // 256 threads = 8 wave32
    rnn_step_wmma<<<grid, block, 0, stream>>>(rates, noise, W, bias,
                                              exp_dt_tau, dt_tau, out);
}